// EmbeddingNet_9380208575078
// MI455X (gfx1250) — compile-verified
//
#include <hip/hip_runtime.h>
#include <hip/hip_bf16.h>

#define AS1 __attribute__((address_space(1)))
#define AS3 __attribute__((address_space(3)))

typedef __attribute__((ext_vector_type(2))) float v2f;
typedef __attribute__((ext_vector_type(8))) float v8f;
typedef int v4i __attribute__((vector_size(16)));   // matches builtin prototype

#if __has_builtin(__builtin_amdgcn_global_load_async_to_lds_b128)
#define HAVE_ASYNC 1
#else
#define HAVE_ASYNC 0
#endif

#if __has_builtin(__builtin_amdgcn_wmma_f32_16x16x4_f32)
#define HAVE_WMMA 1
#else
#define HAVE_WMMA 0
#endif

#if __has_builtin(__builtin_amdgcn_s_wait_asynccnt)
#define WAIT_ASYNC0() __builtin_amdgcn_s_wait_asynccnt(0)
#else
#define WAIT_ASYNC0() asm volatile("s_wait_asynccnt 0x0" ::: "memory")
#endif

constexpr int Bb = 4, Nn = 2048, Mm = 100, Kk = 4, Dd = 5, NT = 40;
constexpr int EMB = 25;                 // D*D
constexpr int ESTR = 32;                // padded emb row stride (floats)
constexpr int OUT_BN = Mm * Kk * EMB;   // 10000 floats per (b,n)
constexpr int JROWS = 128;              // padded to 8 WMMA row tiles

__global__ __launch_bounds__(256) void embednet_kernel(
    const float* __restrict__ x,      // (B,N,M,K)
    const int*   __restrict__ ity,    // (B,N)
    const int*   __restrict__ nbr,    // (B,N,M)
    const float* __restrict__ ntab,   // (NT, D)
    float*       __restrict__ out)    // (B,N,M,K,25)
{
    __shared__ __align__(16) float s_n [NT * Dd];       // 200 floats
    __shared__ __align__(16) float s_x [Mm * Kk];       // 400 floats
    __shared__            float s_jj [JROWS * Dd];      // 640 floats (zero padded)
    __shared__            float s_emb[JROWS * ESTR];    // 128x32 padded (16 KB)

    const int tid = threadIdx.x;
    const int bn  = blockIdx.x;        // 0 .. B*N-1
    const int b   = bn >> 11;          // bn / 2048

    // ---- Stage 0: stage n-table and x-tile into LDS (async on CDNA5) ----
#if HAVE_ASYNC
    if (tid < (NT * Dd) / 4)  // 50 float4
        __builtin_amdgcn_global_load_async_to_lds_b128(
            (AS1 v4i*)(ntab + tid * 4), (AS3 v4i*)(s_n + tid * 4), 0, 0);
    if (tid < (Mm * Kk) / 4)  // 100 float4
        __builtin_amdgcn_global_load_async_to_lds_b128(
            (AS1 v4i*)(x + (size_t)bn * (Mm * Kk) + tid * 4),
            (AS3 v4i*)(s_x + tid * 4), 0, 0);
    WAIT_ASYNC0();
#else
    if (tid < (NT * Dd) / 4)
        *(float4*)(s_n + tid * 4) = *(const float4*)(ntab + tid * 4);
    if (tid < (Mm * Kk) / 4)
        *(float4*)(s_x + tid * 4) =
            *(const float4*)(x + (size_t)bn * (Mm * Kk) + tid * 4);
#endif
    const int it = ity[bn];            // center-atom type (uniform per block)
    __syncthreads();

    // ---- Stage 1: gather neighbor-type embedding rows jj[m][0..4] ----
    if (tid < JROWS) {
        if (tid < Mm) {
            int nb = nbr[(size_t)bn * Mm + tid];
            int jt = (nb == 0) ? 0 : ity[b * Nn + nb - 1];  // cat([0],Imagetype)
            #pragma unroll
            for (int d = 0; d < Dd; ++d) s_jj[tid * Dd + d] = s_n[jt * Dd + d];
        } else {
            #pragma unroll
            for (int d = 0; d < Dd; ++d) s_jj[tid * Dd + d] = 0.0f;
        }
    }
    __syncthreads();

    // ---- Stage 2: emb(100x25) = J(100x5) @ S(5x25), S[q][c]=(c%5==q)?ii[c/5]:0
    //      K=5 split as 4+1 -> two chained V_WMMA_F32_16X16X4_F32 per tile.
    //      D extraction is UNGUARDED: junk cols (25..31) / pad rows (100..127)
    //      land in s_emb padding.
#if HAVE_WMMA
    {
        const int wave = tid >> 5;                 // 8 row tiles of 16 (padded)
        const int lane = tid & 31;
        const int rowA = wave * 16 + (lane & 15);  // A: lanes 0-15 / 16-31 same rows
        const int kh   = (lane >> 4) * 2;          // K slots {0,1} or {2,3}
        v2f A1; A1.x = s_jj[rowA * Dd + kh];
                A1.y = s_jj[rowA * Dd + kh + 1];
        v2f A2; A2.x = (kh == 0) ? s_jj[rowA * Dd + 4] : 0.0f;  // K-slot 0 = jj[.,4]
                A2.y = 0.0f;
        const int rbase = wave * 16 + ((lane >> 4) ? 8 : 0);    // D rows rbase..+7
        #pragma unroll
        for (int ct = 0; ct < 2; ++ct) {           // col tiles: 0-15, 16-24(+junk)
            const int  col = ct * 16 + (lane & 15);
            const bool cv  = (col < EMB);
            const float iv = cv ? s_n[it * Dd + col / Dd] : 0.0f;  // ii[d1]
            const int  d2  = col % Dd;             // junk cols: iv==0 anyway
            // B layout: vgpr0 holds rows K=kh, vgpr1 rows K=kh+1 (cols = lane&15)
            v2f B1; B1.x = (d2 == kh)     ? iv : 0.0f;
                    B1.y = (d2 == kh + 1) ? iv : 0.0f;
            v2f B2; B2.x = (d2 == 4 && kh == 0) ? iv : 0.0f;
                    B2.y = 0.0f;
            v8f acc = {};
            acc = __builtin_amdgcn_wmma_f32_16x16x4_f32(
                      false, A2, false, B2, (short)0, acc, false, false);
            acc = __builtin_amdgcn_wmma_f32_16x16x4_f32(
                      false, A1, false, B1, (short)0, acc, false, false);
            // D layout: vgpr i holds rows (rbase+i); immediate-offset LDS stores
            float* p = &s_emb[rbase * ESTR + col];
            #pragma unroll
            for (int i = 0; i < 8; ++i) p[i * ESTR] = acc[i];
        }
    }
#else
    for (int p = tid; p < Mm * EMB; p += 256) {
        int m = p / EMB, j = p - m * EMB;
        s_emb[m * ESTR + j] = s_n[it * Dd + j / Dd] * s_jj[m * Dd + j % Dd];
    }
#endif
    __syncthreads();

    // ---- Stage 3: out[m][k*25+j] = x[m][k] * emb[m][j], float4 stores ----
    // Thread owns fixed float4-phase tq in [0,25); index math hoisted.
    if (tid < 250) {
        const int tq = tid % EMB;      // float4 slot within 100-float m-block
        const int tm = tid / EMB;      // 0..9
        const int e  = 4 * tq;
        const int k0 = (e    ) / EMB, j0 = (e    ) % EMB;
        const int k1 = (e + 1) / EMB, j1 = (e + 1) % EMB;
        const int k2 = (e + 2) / EMB, j2 = (e + 2) % EMB;
        const int k3 = (e + 3) / EMB, j3 = (e + 3) % EMB;
        float4* o4 = (float4*)(out + (size_t)bn * OUT_BN);
        #pragma unroll 2
        for (int m = tm; m < Mm; m += 10) {
            const float* em = &s_emb[m * ESTR];
            const float* xm = &s_x[m * Kk];
            float4 v;
            v.x = xm[k0] * em[j0];
            v.y = xm[k1] * em[j1];
            v.z = xm[k2] * em[j2];
            v.w = xm[k3] * em[j3];
            o4[m * EMB + tq] = v;      // contiguous 400B runs per 25 lanes
        }
    }
}

extern "C" void kernel_launch(void* const* d_in, const int* in_sizes, int n_in,
                              void* d_out, int out_size, void* d_ws, size_t ws_size,
                              hipStream_t stream) {
    const float* x    = (const float*)d_in[0];   // (B,N,M,K) fp32
    const int*   ity  = (const int*)d_in[1];     // (B,N)
    const int*   nbr  = (const int*)d_in[2];     // (B,N,M)
    const float* ntab = (const float*)d_in[3];   // (40,5) fp32
    float* out = (float*)d_out;                  // (B,N,M,K,25) fp32

    dim3 grid(Bb * Nn);   // 8192 blocks, one per (b,n)
    dim3 block(256);      // 8 waves (wave32)
    hipLaunchKernelGGL(embednet_kernel, grid, block, 0, stream,
                       x, ity, nbr, ntab, out);
}